// MixUniformAffineQuantizer_85160611545365
// MI455X (gfx1250) — compile-verified
//
#include <hip/hip_runtime.h>

#define ROWS 4096
#define COLS 11008
#define GSZ  128
#define NB   (COLS / GSZ)            // 86 blocks per row
#define LPG  4                       // lanes per group
#define EPL  (GSZ / LPG)             // 32 elements per lane
#define GPW  8                       // groups per wave per step (32/LPG)
#define WPB  4                       // waves per 128-thread block
#define NSTEP 4                      // steps per wave (double-buffered LDS)
#define GPB  (GPW * WPB * NSTEP)     // 128 groups per block
#define NBLOCKS ((ROWS * NB) / GPB)  // 352256/128 = 2752, exact
#define PADF 132                     // padded floats per group in LDS (528B, 16B aligned)

typedef float v4f __attribute__((ext_vector_type(4)));
typedef int   v4i __attribute__((vector_size(4 * sizeof(int))));

#if defined(__has_builtin)
#  if __has_builtin(__builtin_amdgcn_global_load_async_to_lds_b128) && \
      __has_builtin(__builtin_amdgcn_s_wait_asynccnt)
#    define USE_ASYNC_LDS 1
#  endif
#  if __has_builtin(__builtin_amdgcn_ds_swizzle)
#    define USE_SWIZZLE 1
#  endif
#endif
#ifndef USE_ASYNC_LDS
#  define USE_ASYNC_LDS 0
#endif
#ifndef USE_SWIZZLE
#  define USE_SWIZZLE 0
#endif

#define AS1 __attribute__((address_space(1)))
#define AS3 __attribute__((address_space(3)))

#if USE_ASYNC_LDS
__device__ __forceinline__ void async_group_load(const float* src, float* lds_dst) {
    AS3 void* lp = (AS3 void*)lds_dst;
    __builtin_amdgcn_global_load_async_to_lds_b128(
        (AS1 v4i*)(unsigned long long)src, (AS3 v4i*)lp, 0, 0);
}
#endif

// xor-butterfly exchange within the 4-lane quad, zero VALU overhead via ds_swizzle
#if USE_SWIZZLE
#define SWZ(v, imm) __int_as_float(__builtin_amdgcn_ds_swizzle(__float_as_int(v), (imm)))
#define XRED(v, OP) do {                      \
    v = OP(v, SWZ(v, 0x041F));  /* xor 1 */   \
    v = OP(v, SWZ(v, 0x081F));  /* xor 2 */   \
} while (0)
#else
#define XRED(v, OP) do {                      \
    v = OP(v, __shfl_xor(v, 1, 32));          \
    v = OP(v, __shfl_xor(v, 2, 32));          \
} while (0)
#endif

__device__ __forceinline__ float fadd(float a, float b) { return a + b; }
__device__ __forceinline__ v4f vmin4(v4f a, v4f b) {
    v4f r; r.x = fminf(a.x,b.x); r.y = fminf(a.y,b.y); r.z = fminf(a.z,b.z); r.w = fminf(a.w,b.w); return r;
}
__device__ __forceinline__ v4f vmax4(v4f a, v4f b) {
    v4f r; r.x = fmaxf(a.x,b.x); r.y = fmaxf(a.y,b.y); r.z = fmaxf(a.z,b.z); r.w = fmaxf(a.w,b.w); return r;
}
__device__ __forceinline__ v4f vabs4(v4f a) {
    v4f r; r.x = fabsf(a.x); r.y = fabsf(a.y); r.z = fabsf(a.z); r.w = fabsf(a.w); return r;
}

// rint(e/scale) via Markstein-refined reciprocal quotient:
// inv = RN(1/scale); q1 = q0 + (e - scale*q0)*inv is within ~0.5 ulp of e/scale.
__device__ __forceinline__ float qround(float e, float scale, float inv) {
    float q0 = e * inv;
    float q1 = fmaf(fmaf(-scale, q0, e), inv, q0);
    return rintf(q1);
}

__global__ __launch_bounds__(WPB * 32)
void MixUniformAffineQuantizer_kernel(const float* __restrict__ x,
                                      const float* __restrict__ up,
                                      const float* __restrict__ low,
                                      float* __restrict__ out)
{
    const int lane  = threadIdx.x & 31;
    // force wave id into an SGPR so per-step base arithmetic stays scalar
    const int wave  = __builtin_amdgcn_readfirstlane((int)(threadIdx.x >> 5));
    const int j     = lane >> 2;         // which of the 8 groups this lane serves
    const int c     = lane & 3;          // 128B chunk within the group
    const int wbase = blockIdx.x * GPB + wave * (GPW * NSTEP);

#if USE_ASYNC_LDS
    __shared__ __align__(16) float sbuf[WPB][2][GPW * PADF];  // ~33.8 KB
#pragma unroll
    for (int jj = 0; jj < GPW; ++jj)     // preload step 0 (coalesced 512B per instr)
        async_group_load(x + (size_t)(wbase + jj) * GSZ + lane * 4,
                         &sbuf[wave][0][jj * PADF + lane * 4]);
#endif

#pragma unroll
    for (int s = 0; s < NSTEP; ++s) {
        const int base = wbase + s * GPW;     // scalar
        const int g = base + j;               // flat group id == (row*NB + blk)
        v4f d[8];

#if USE_ASYNC_LDS
        if (s + 1 < NSTEP) {
#pragma unroll
            for (int jj = 0; jj < GPW; ++jj)
                async_group_load(x + (size_t)(base + GPW + jj) * GSZ + lane * 4,
                                 &sbuf[wave][(s + 1) & 1][jj * PADF + lane * 4]);
            __builtin_amdgcn_s_wait_asynccnt(GPW);  // step s landed (in-order)
        } else {
            __builtin_amdgcn_s_wait_asynccnt(0);
        }
        {
            const float* lp = &sbuf[wave][s & 1][j * PADF + c * EPL];
#pragma unroll
            for (int k = 0; k < 8; ++k) d[k] = *(const v4f*)(lp + k * 4);  // ds_load_b128
        }
#else
#pragma unroll
        for (int k = 0; k < 8; ++k)
            d[k] = ((const v4f*)x)[(size_t)g * 32 + c * 8 + k];
#endif

        // ---- intra-thread reduce over 32 elements (each instr serves 8 groups) ----
        v4f tmn = vmin4(vmin4(vmin4(d[0], d[1]), vmin4(d[2], d[3])),
                        vmin4(vmin4(d[4], d[5]), vmin4(d[6], d[7])));
        v4f tmx = vmax4(vmax4(vmax4(d[0], d[1]), vmax4(d[2], d[3])),
                        vmax4(vmax4(d[4], d[5]), vmax4(d[6], d[7])));
        v4f tsm = ((d[0] + d[1]) + (d[2] + d[3])) + ((d[4] + d[5]) + (d[6] + d[7]));
        v4f tas = ((vabs4(d[0]) + vabs4(d[1])) + (vabs4(d[2]) + vabs4(d[3]))) +
                  ((vabs4(d[4]) + vabs4(d[5])) + (vabs4(d[6]) + vabs4(d[7])));
        float mn = fminf(fminf(tmn.x, tmn.y), fminf(tmn.z, tmn.w));
        float mx = fmaxf(fmaxf(tmx.x, tmx.y), fmaxf(tmx.z, tmx.w));
        float sm = (tsm.x + tsm.y) + (tsm.z + tsm.w);
        float as = (tas.x + tas.y) + (tas.z + tas.w);

        // ---- quad (4-lane) butterfly: 2 stages, lanes stay within their group ----
        XRED(mn, fminf);
        XRED(mx, fmaxf);
        XRED(sm, fadd);
        XRED(as, fadd);

        // ---- per-group scalars (each instr serves 8 groups) ----
        const float u = up[g];
        const float l = low[g];
        const float su = 1.0f / (1.0f + expf(-u));
        const float sl = 1.0f / (1.0f + expf(-l));

        const int sblk = base % NB;           // scalar mod (base is wave-uniform)
        int blk = sblk + j;                   // j <= 7, sblk <= 85 -> one wrap max
        blk -= (blk >= NB) ? NB : 0;
        const bool ternary = (blk == 0);

        float levels;
        {
            const int p = (blk - 1) % 5;         // _PREC pattern [2,3,4,3,2]
            const int prec = (p == 2) ? 4 : ((p == 1 || p == 3) ? 3 : 2);
            levels = ternary ? 1.0f : (float)((1 << prec) - 1);
        }

        const float mean   = sm * (1.0f / GSZ);
        const float amean  = as * (1.0f / GSZ);
        const float xmaxs  = su * mx;
        const float xmins  = sl * mn;
        const float scale_r = (xmaxs - xmins) / levels;   // unclipped, feeds zp
        const float scale_t = amean * (su + 0.5f);

        float scale = ternary ? scale_t : scale_r;
        scale = fminf(fmaxf(scale, 1e-5f), 1e4f);

        float zp = ternary ? mean : (-xmins / scale_r);
        zp = fminf(fmaxf(zp, -1e4f), 1e4f);
        if (!ternary) zp = rintf(zp);            // round-half-even, matches jnp.round

        // ---- dequantize this thread's 32 elements ----
        if (!ternary) {
            const float inv = 1.0f / scale;      // one exact division per group
            const float lo  = -zp;               // clip(r+zp,0,levels)-zp == clamp(r,lo,hi)
            const float hi  = levels - zp;       // all integers < 2^24: exact
#pragma unroll
            for (int k = 0; k < 8; ++k) {
                v4f o;
                o.x = fminf(fmaxf(qround(d[k].x, scale, inv), lo), hi) * scale;
                o.y = fminf(fmaxf(qround(d[k].y, scale, inv), lo), hi) * scale;
                o.z = fminf(fmaxf(qround(d[k].z, scale, inv), lo), hi) * scale;
                o.w = fminf(fmaxf(qround(d[k].w, scale, inv), lo), hi) * scale;
                d[k] = o;
            }
        } else {   // 1.2% of groups; block usually skipped via execz
#pragma unroll
            for (int k = 0; k < 8; ++k) {
                v4f o;
                o.x = (float)((d[k].x > zp) - (d[k].x < zp)) * scale;
                o.y = (float)((d[k].y > zp) - (d[k].y < zp)) * scale;
                o.z = (float)((d[k].z > zp) - (d[k].z < zp)) * scale;
                o.w = (float)((d[k].w > zp) - (d[k].w < zp)) * scale;
                d[k] = o;
            }
        }

        v4f* op = (v4f*)out + (size_t)g * 32 + c * 8;
#pragma unroll
        for (int k = 0; k < 8; ++k)
            __builtin_nontemporal_store(d[k], op + k);
    }
}

extern "C" void kernel_launch(void* const* d_in, const int* in_sizes, int n_in,
                              void* d_out, int out_size, void* d_ws, size_t ws_size,
                              hipStream_t stream) {
    const float* x   = (const float*)d_in[0];
    const float* up  = (const float*)d_in[1];
    const float* low = (const float*)d_in[2];
    float* out = (float*)d_out;
    (void)in_sizes; (void)n_in; (void)out_size; (void)d_ws; (void)ws_size;
    MixUniformAffineQuantizer_kernel<<<NBLOCKS, WPB * 32, 0, stream>>>(x, up, low, out);
}